// CRF_8469675508122
// MI455X (gfx1250) — compile-verified
//
#include <hip/hip_runtime.h>

#define BB 2048
#define TT 512
#define NN 32

typedef float v2f __attribute__((ext_vector_type(2)));
typedef float v8f __attribute__((ext_vector_type(8)));

#define LOG2E 1.4426950408889634f
#define LN2   0.6931471805599453f

// Raw transcendentals: inputs are never denormal here (exp args <= 0,
// log args in [0.9, ~35]), so skip libm's range-fixup code.
__device__ __forceinline__ float fexp(float x) {
  return __builtin_amdgcn_exp2f(x * LOG2E);
}
__device__ __forceinline__ float flog(float x) {
  return __builtin_amdgcn_logf(x) * LN2;
}

__device__ __forceinline__ float half_max16(float v) {
  v = fmaxf(v, __shfl_xor(v, 1, 32));
  v = fmaxf(v, __shfl_xor(v, 2, 32));
  v = fmaxf(v, __shfl_xor(v, 4, 32));
  v = fmaxf(v, __shfl_xor(v, 8, 32));
  return v;
}
__device__ __forceinline__ float half_sum16(float v) {
  v += __shfl_xor(v, 1, 32);
  v += __shfl_xor(v, 2, 32);
  v += __shfl_xor(v, 4, 32);
  v += __shfl_xor(v, 8, 32);
  return v;
}

// One wave (32 lanes) per 16 batch rows. Alpha kept in WMMA C/D layout:
//   VGPR r: lanes 0-15 -> row r,   col = lane
//           lanes 16-31 -> row r+8, col = lane-16
__global__ void __launch_bounds__(32) crf_fwd(
    const float* __restrict__ em,    // [B,T,N]
    const int*   __restrict__ tags,  // [B,T]
    const float* __restrict__ mask,  // [B,T]
    const float* __restrict__ trans, // [N,N]
    const float* __restrict__ stt,   // [N]
    const float* __restrict__ ent,   // [N]
    float* __restrict__ ws)          // [B] per-row (log_z - gold)
{
  __shared__ float lds_v[16 * 34];   // V tile, pad 34 (even -> b64 aligned)
  __shared__ float lds_tr[NN * NN];  // raw transitions for gold gather

  const int lane = threadIdx.x;
  const int m    = lane & 15;
  const int half = lane >> 4;
  const int b0   = blockIdx.x * 16;

  for (int i = lane; i < NN * NN; i += 32) lds_tr[i] = trans[i];

  // B operands: exp(transitions), ISA B layout (K striped: VGPR0={K0|K2}, VGPR1={K1|K3})
  v2f Bt0[8], Bt1[8];
#pragma unroll
  for (int k = 0; k < 8; ++k) {
    const int kr = 4 * k + 2 * half;
    Bt0[k].x = fexp(trans[kr * NN + m]);
    Bt0[k].y = fexp(trans[(kr + 1) * NN + m]);
    Bt1[k].x = fexp(trans[kr * NN + 16 + m]);
    Bt1[k].y = fexp(trans[(kr + 1) * NN + 16 + m]);
  }

  const float st0 = stt[m], st1 = stt[16 + m];
  const float et0 = ent[m], et1 = ent[16 + m];

  // Per-lane 32-bit element offsets (base pointer advances uniformly per step).
  int ro0[8], ro1[8];
#pragma unroll
  for (int r = 0; r < 8; ++r) {
    const int row = b0 + r + 8 * half;
    ro0[r] = row * TT * NN + m;
    ro1[r] = ro0[r] + 16;
  }
  const int rowT = (b0 + m) * TT;       // tags row offset (elements)
  const int roG  = (b0 + m) * TT * NN;  // gold-gather row offset (elements)

  // sequence length of row (b0+m) (mask is a prefix of ones)
  float lenrow = 0.f;
  {
    const float4* mp = reinterpret_cast<const float4*>(mask + (long)(b0 + m) * TT);
    for (int t4 = 0; t4 < TT / 4; ++t4) {
      float4 q = mp[t4];
      lenrow += q.x + q.y + q.z + q.w;
    }
  }
  float lenv[8];
#pragma unroll
  for (int r = 0; r < 8; ++r) lenv[r] = __shfl(lenrow, r + 8 * half, 32);

  __syncthreads();

  // alpha0 = start + emissions[:,0,:]
  v8f a0, a1;
#pragma unroll
  for (int r = 0; r < 8; ++r) {
    a0[r] = st0 + em[ro0[r]];
    a1[r] = st1 + em[ro1[r]];
  }

  // gold score: lane handles row b0+m (upper half duplicates lanes 0-15)
  int   ptag    = tags[rowT];
  float gold    = stt[ptag] + em[roG + ptag];
  int   lasttag = ptag;

  // ---- software pipeline: preload emissions + tag for t = 1 ----
  float e0c[8], e1c[8];
  {
    const float* emt = em + NN;
#pragma unroll
    for (int r = 0; r < 8; ++r) { e0c[r] = emt[ro0[r]]; e1c[r] = emt[ro1[r]]; }
  }
  int tgc = tags[rowT + 1];

  for (int t = 1; t < TT; ++t) {
    const float tf = (float)t;
    const int   tn = (t < TT - 1) ? t + 1 : t;   // uniform (SALU)

    // ---- issue next-step loads early (independent of alpha) ----
    float e0n[8], e1n[8];
    {
      const float* emn = em + (long)tn * NN;
#pragma unroll
      for (int r = 0; r < 8; ++r) { e0n[r] = emn[ro0[r]]; e1n[r] = emn[ro1[r]]; }
    }
    const int tgn = tags[rowT + tn];

    // ---- gold contribution (gather line is warm from last step's loads) ----
    {
      const float c  = lds_tr[ptag * NN + tgc] + em[roG + t * NN + tgc];
      const bool  on = tf < lenrow;
      gold    += on ? c : 0.f;
      lasttag  = on ? tgc : lasttag;
      ptag     = tgc;
    }

    // ---- per-row max, V = exp(alpha - m) -> LDS (A-layout friendly) ----
    float mx[8];
#pragma unroll
    for (int r = 0; r < 8; ++r) {
      mx[r] = half_max16(fmaxf(a0[r], a1[r]));
      const int row = r + 8 * half;
      lds_v[row * 34 + m]      = fexp(a0[r] - mx[r]);
      lds_v[row * 34 + 16 + m] = fexp(a1[r] - mx[r]);
    }
    asm volatile("s_wait_dscnt 0" ::: "memory");

    // ---- D = V(16x32) @ expT(32x32): 2 col-tiles, 2x 4-deep WMMA chains each ----
    v8f d0a = {0.f, 0.f, 0.f, 0.f, 0.f, 0.f, 0.f, 0.f};
    v8f d0b = d0a, d1a = d0a, d1b = d0a;
#pragma unroll
    for (int k = 0; k < 4; ++k) {
      const v2f avA = *reinterpret_cast<const v2f*>(&lds_v[m * 34 + 4 * k + 2 * half]);
      const v2f avB = *reinterpret_cast<const v2f*>(&lds_v[m * 34 + 4 * (k + 4) + 2 * half]);
      d0a = __builtin_amdgcn_wmma_f32_16x16x4_f32(false, avA, false, Bt0[k],
                                                  (short)0, d0a, false, false);
      d1a = __builtin_amdgcn_wmma_f32_16x16x4_f32(false, avA, false, Bt1[k],
                                                  (short)0, d1a, false, false);
      d0b = __builtin_amdgcn_wmma_f32_16x16x4_f32(false, avB, false, Bt0[k + 4],
                                                  (short)0, d0b, false, false);
      d1b = __builtin_amdgcn_wmma_f32_16x16x4_f32(false, avB, false, Bt1[k + 4],
                                                  (short)0, d1b, false, false);
    }
    const v8f d0 = d0a + d0b;
    const v8f d1 = d1a + d1b;

    // ---- new_alpha = e + m + log(dot), masked update ----
#pragma unroll
    for (int r = 0; r < 8; ++r) {
      const float na0 = e0c[r] + mx[r] + flog(d0[r]);
      const float na1 = e1c[r] + mx[r] + flog(d1[r]);
      const bool upd = tf < lenv[r];
      a0[r] = upd ? na0 : a0[r];
      a1[r] = upd ? na1 : a1[r];
    }

    // ---- rotate pipeline ----
#pragma unroll
    for (int r = 0; r < 8; ++r) { e0c[r] = e0n[r]; e1c[r] = e1n[r]; }
    tgc = tgn;
  }

  gold += ent[lasttag];

  // log_z = logsumexp(alpha + end_trans) per row; write log_z - gold
#pragma unroll
  for (int r = 0; r < 8; ++r) {
    const float z0 = a0[r] + et0;
    const float z1 = a1[r] + et1;
    const float mm = half_max16(fmaxf(z0, z1));
    const float ss = half_sum16(fexp(z0 - mm) + fexp(z1 - mm));
    const float logz = mm + flog(ss);
    const float g = __shfl(gold, r + 8 * half, 32);
    if (m == 0) ws[b0 + r + 8 * half] = logz - g;
  }
}

// Deterministic mean over B values (fixed partition + tree -> replay-stable).
__global__ void __launch_bounds__(256) crf_reduce(const float* __restrict__ ws,
                                                  float* __restrict__ out) {
  __shared__ float sm[256];
  float s = 0.f;
  for (int i = threadIdx.x; i < BB; i += 256) s += ws[i];
  sm[threadIdx.x] = s;
  __syncthreads();
  for (int k = 128; k > 0; k >>= 1) {
    if (threadIdx.x < (unsigned)k) sm[threadIdx.x] += sm[threadIdx.x + k];
    __syncthreads();
  }
  if (threadIdx.x == 0) out[0] = sm[0] / (float)BB;
}

extern "C" void kernel_launch(void* const* d_in, const int* in_sizes, int n_in,
                              void* d_out, int out_size, void* d_ws, size_t ws_size,
                              hipStream_t stream) {
  const float* em    = (const float*)d_in[0];
  const int*   tags  = (const int*)d_in[1];
  const float* mask  = (const float*)d_in[2];
  const float* trans = (const float*)d_in[3];
  const float* stt   = (const float*)d_in[4];
  const float* ent   = (const float*)d_in[5];
  float* ws = (float*)d_ws;

  crf_fwd<<<BB / 16, 32, 0, stream>>>(em, tags, mask, trans, stt, ent, ws);
  crf_reduce<<<1, 256, 0, stream>>>(ws, (float*)d_out);
}